// TokenConditioner_23811298689912
// MI455X (gfx1250) — compile-verified
//
#include <hip/hip_runtime.h>
#include <math.h>

typedef __attribute__((ext_vector_type(2))) float v2f;
typedef __attribute__((ext_vector_type(8))) float v8f;

#define B_   16
#define TIN  57
#define D_   256
#define C_   64
#define TOUT 113
#define FB   1025
#define NPAD 128            // TOUT padded to 8 N-tiles of 16

// ---------------------------------------------------------------------------
// Kernel 1: noise embedding  e[b][c] = silu(0.25*log(sigma_b)*w1 + b1) @ w2 + b2
// ---------------------------------------------------------------------------
__global__ void ne_kernel(const float* __restrict__ sigma,
                          const float* __restrict__ w1, const float* __restrict__ b1,
                          const float* __restrict__ w2, const float* __restrict__ b2,
                          float* __restrict__ e) {
    int b = blockIdx.x;          // 0..15
    int c = threadIdx.x;         // 0..63
    float cn = 0.25f * __logf(sigma[b]);
    float acc = b2[c];
    #pragma unroll 8
    for (int j = 0; j < 128; ++j) {
        float t = cn * w1[j] + b1[j];
        float s = t / (1.0f + __expf(-t));     // SiLU
        acc += s * w2[c * 128 + j];
    }
    e[b * C_ + c] = acc;
}

// ---------------------------------------------------------------------------
// Kernel 2: projection GEMM  h[m][c] = z[m][:] @ proj_w[c][:] + proj_b[c]
// m = b*TIN + t  (M=912), N=64, K=256.  One wave per 16x16 tile, f32 WMMA.
// ---------------------------------------------------------------------------
__global__ __launch_bounds__(32)
void proj_kernel(const float* __restrict__ z, const float* __restrict__ pw,
                 const float* __restrict__ pb, float* __restrict__ h) {
    const int lane = threadIdx.x;           // 0..31, full wave, EXEC all ones
    const int mt = blockIdx.x;              // 0..56
    const int nt = blockIdx.y;              // 0..3
    const int mrow = mt * 16 + (lane & 15); // A row this lane supplies
    const int ncol = nt * 16 + (lane & 15); // B col this lane supplies
    const int kb   = (lane >> 4) * 2;       // K sub-offset (0 or 2)

    const float* za = z  + mrow * D_;
    const float* bw = pw + ncol * D_;       // proj_w[n][k], row-major (C,D)

    v8f acc = {};
    #pragma unroll 8
    for (int k0 = 0; k0 < D_; k0 += 4) {
        v2f a, bf;
        a[0]  = za[k0 + kb];
        a[1]  = za[k0 + kb + 1];
        bf[0] = bw[k0 + kb];
        bf[1] = bw[k0 + kb + 1];
        acc = __builtin_amdgcn_wmma_f32_16x16x4_f32(false, a, false, bf,
                                                    (short)0, acc, false, false);
    }

    float bias = pb[ncol];
    const int mbase = mt * 16 + 8 * (lane >> 4);
    #pragma unroll
    for (int r = 0; r < 8; ++r) {
        h[(mbase + r) * C_ + ncol] = acc[r] + bias;
    }
}

// ---------------------------------------------------------------------------
// Kernel 3: fused resample(57->113) + noise-embed add + 1x1 conv, WMMA f32.
// Per batch: y[o][t] = sum_c conv_w[o][c] * (interp(h)[c][t] + e[c]) + conv_b[o]
// M=64 (o), K=64 (c), N padded to 128 (t). Grid (b, mt, nt), one wave/tile.
// ---------------------------------------------------------------------------
__global__ __launch_bounds__(32)
void conv_kernel(const float* __restrict__ h, const float* __restrict__ e,
                 const float* __restrict__ cw, const float* __restrict__ cb,
                 float* __restrict__ y) {
    const int lane = threadIdx.x;
    const int b  = blockIdx.x;              // 0..15
    const int mt = blockIdx.y;              // 0..3
    const int nt = blockIdx.z;              // 0..7
    const int l15 = lane & 15;
    const int kb  = (lane >> 4) * 2;

    // interp coordinates for this lane's B-column (time index); hoisted from K loop
    const int t  = nt * 16 + l15;           // 0..127 (>=113 are pad columns)
    const int tc = t < TOUT ? t : (TOUT - 1);
    float pos = ((float)tc + 0.5f) * ((float)TIN / (float)TOUT) - 0.5f;
    pos = fminf(fmaxf(pos, 0.0f), (float)(TIN - 1));
    int   i0 = (int)floorf(pos);
    int   i1 = i0 + 1 < TIN ? i0 + 1 : TIN - 1;
    float w  = pos - (float)i0;

    const float* h0 = h + (b * TIN + i0) * C_;   // row-major [m][c]
    const float* h1 = h + (b * TIN + i1) * C_;
    const float* eb = e + b * C_;
    const float* aw = cw + (mt * 16 + l15) * C_; // conv_w[o][c]

    v8f acc = {};
    #pragma unroll 4
    for (int k0 = 0; k0 < C_; k0 += 4) {
        int c0 = k0 + kb;
        v2f a, bf;
        a[0] = aw[c0];
        a[1] = aw[c0 + 1];
        float p0 = h0[c0],     q0 = h1[c0];
        float p1 = h0[c0 + 1], q1 = h1[c0 + 1];
        bf[0] = p0 + (q0 - p0) * w + eb[c0];
        bf[1] = p1 + (q1 - p1) * w + eb[c0 + 1];
        acc = __builtin_amdgcn_wmma_f32_16x16x4_f32(false, a, false, bf,
                                                    (short)0, acc, false, false);
    }

    if (t < TOUT) {
        const int obase = mt * 16 + 8 * (lane >> 4);
        #pragma unroll
        for (int r = 0; r < 8; ++r) {
            int o = obase + r;
            y[(b * C_ + o) * TOUT + t] = acc[r] + cb[o];
        }
    }
}

// ---------------------------------------------------------------------------
// Kernel 4: broadcast y[B,C,TOUT] over F_BINS -> out[B,C,FB,TOUT]. 474 MB of
// stores; this is the whole runtime. Row cached in LDS (duplicated +3 to kill
// wrap branches), body written as 16B global_store_b128, one %113 per 16 B.
// ---------------------------------------------------------------------------
__global__ __launch_bounds__(256)
void bcast_kernel(const float* __restrict__ y, float* __restrict__ out) {
    __shared__ float row[TOUT + 3];
    const unsigned bc = blockIdx.x;                    // 0..1023
    const float* yr = y + bc * TOUT;
    for (int i = threadIdx.x; i < TOUT + 3; i += blockDim.x)
        row[i] = yr[i < TOUT ? i : i - TOUT];
    __syncthreads();

    const unsigned f0 = (blockIdx.y * FB) / gridDim.y;
    const unsigned f1 = ((blockIdx.y + 1) * FB) / gridDim.y;
    const unsigned chunk = bc * (unsigned)(FB * TOUT);  // float offset of (b,c) chunk
    const unsigned Gs = chunk + f0 * TOUT;
    const unsigned Ge = chunk + f1 * TOUT;
    unsigned Ga = (Gs + 3u) & ~3u;                      // first 16B-aligned float
    if (Ga > Ge) Ga = Ge;
    const unsigned Gt = Ga + ((Ge - Ga) & ~3u);         // end of vector region

    for (unsigned g = Gs + threadIdx.x; g < Ga; g += blockDim.x)
        out[g] = row[(g - chunk) % (unsigned)TOUT];

    for (unsigned g = Ga + threadIdx.x * 4u; g < Gt; g += blockDim.x * 4u) {
        unsigned r = (g - chunk) % (unsigned)TOUT;      // row is duplicated: no wrap
        float4 v = make_float4(row[r], row[r + 1], row[r + 2], row[r + 3]);
        *reinterpret_cast<float4*>(out + g) = v;
    }

    for (unsigned g = Gt + threadIdx.x; g < Ge; g += blockDim.x)
        out[g] = row[(g - chunk) % (unsigned)TOUT];
}

// ---------------------------------------------------------------------------
extern "C" void kernel_launch(void* const* d_in, const int* in_sizes, int n_in,
                              void* d_out, int out_size, void* d_ws, size_t ws_size,
                              hipStream_t stream) {
    const float* z      = (const float*)d_in[0];
    const float* sigma  = (const float*)d_in[1];
    const float* proj_w = (const float*)d_in[2];
    const float* proj_b = (const float*)d_in[3];
    const float* conv_w = (const float*)d_in[4];
    const float* conv_b = (const float*)d_in[5];
    const float* ne_w1  = (const float*)d_in[6];
    const float* ne_b1  = (const float*)d_in[7];
    const float* ne_w2  = (const float*)d_in[8];
    const float* ne_b2  = (const float*)d_in[9];
    float* out = (float*)d_out;

    float* e = (float*)d_ws;                 // B*C          = 1024 floats
    float* h = e + B_ * C_;                  // B*TIN*C      = 58368 floats
    float* y = h + B_ * TIN * C_;            // B*C*TOUT     = 115712 floats

    ne_kernel<<<dim3(B_), dim3(C_), 0, stream>>>(sigma, ne_w1, ne_b1, ne_w2, ne_b2, e);
    proj_kernel<<<dim3(TIN, C_ / 16), dim3(32), 0, stream>>>(z, proj_w, proj_b, h);
    conv_kernel<<<dim3(B_, C_ / 16, NPAD / 16), dim3(32), 0, stream>>>(h, e, conv_w, conv_b, y);
    bcast_kernel<<<dim3(B_ * C_, 32), dim3(256), 0, stream>>>(y, out);
}